// DGLTemporalGAT_23922967839173
// MI455X (gfx1250) — compile-verified
//
#include <hip/hip_runtime.h>
#include <math.h>

// GATv2 dense attention on MI455X (gfx1250, wave32).
// B=64, W=128, F=64, H=2, OUTF=64.  Inputs f32, output f32 [B,W,OUTF].

typedef __attribute__((ext_vector_type(2))) float v2f;
typedef __attribute__((ext_vector_type(8))) float v8f;

#define B_    64
#define W_    128
#define F_    64
#define HF_   128    // H*OUTF
#define OUTF_ 64
#define ROWS_ (B_ * W_)   // 8192

__device__ __forceinline__ v8f wmma_f32_16x16x4(v2f a, v2f b, v8f c) {
    // D = A(16x4 f32) x B(4x16 f32) + C(16x16 f32)
    return __builtin_amdgcn_wmma_f32_16x16x4_f32(
        /*neg_a=*/false, a, /*neg_b=*/false, b,
        /*c_mod=*/(short)0, c, /*reuse_a=*/false, /*reuse_b=*/false);
}

// ---------------------------------------------------------------------------
// Kernel 1: fsrc = x @ w_src^T + b_src ; fdst = x @ w_dst^T + b_dst
// x: [8192,64], w: [128,64] row-major, out: [8192,128]
// Grid: 128 blocks x 256 threads. Block handles 64 rows; 8 waves x 8 tasks
// (task = {4 M-tiles} x {8 N-tiles} x {src,dst}).
// Row pad = 68 floats: 16B-aligned b128 LDS ops; column reads hit banks
// (4m + k) % 64 -> conflict-free across all 32 lanes.
// ---------------------------------------------------------------------------
__global__ __launch_bounds__(256) void gat_proj_kernel(
    const float* __restrict__ x,
    const float* __restrict__ w_src, const float* __restrict__ b_src,
    const float* __restrict__ w_dst, const float* __restrict__ b_dst,
    float* __restrict__ fsrc, float* __restrict__ fdst)
{
    __shared__ float xs[64][68];
    __shared__ float ws[128][68];
    __shared__ float wd[128][68];

    const int tid  = threadIdx.x;
    const int row0 = blockIdx.x * 64;

    // Stage x tile: 64 rows x 16 float4 (ds_store_b128)
    for (int v = tid; v < 64 * 16; v += 256) {
        const int r = v >> 4, c4 = (v & 15) << 2;
        *(float4*)&xs[r][c4] = *(const float4*)(x + (size_t)(row0 + r) * F_ + c4);
    }
    // Stage both weight matrices: 128 rows x 16 float4 each
    for (int v = tid; v < 128 * 16; v += 256) {
        const int r = v >> 4, c4 = (v & 15) << 2;
        *(float4*)&ws[r][c4] = *(const float4*)(w_src + (size_t)r * F_ + c4);
        *(float4*)&wd[r][c4] = *(const float4*)(w_dst + (size_t)r * F_ + c4);
    }
    __syncthreads();

    const int wave = tid >> 5;
    const int lane = tid & 31;
    const int m    = lane & 15;            // M (A) or N (B/C) index in tile
    const int koff = (lane >> 4) << 1;     // 0 or 2 (A/B K split per half-wave)
    const int rb   = (lane >> 4) << 3;     // C layout: +8 rows for lanes 16-31

    for (int t = 0; t < 8; ++t) {
        const int task = wave * 8 + t;     // [0,64)
        const int mt   = task & 3;         // M tile (of 4)
        const int nt   = (task >> 2) & 7;  // N tile (of 8)
        const int mat  = task >> 5;        // 0=src 1=dst
        const float (*wsh)[68] = mat ? wd : ws;
        const float* bias      = mat ? b_dst : b_src;
        float* outp            = mat ? fdst : fsrc;

        v8f acc = {};
        #pragma unroll
        for (int k0 = 0; k0 < F_; k0 += 4) {
            v2f a, b;
            a.x = xs[mt * 16 + m][k0 + koff];
            a.y = xs[mt * 16 + m][k0 + koff + 1];
            // B[k][n] = w[n][k]  (transposed read, padded rows)
            b.x = wsh[nt * 16 + m][k0 + koff];
            b.y = wsh[nt * 16 + m][k0 + koff + 1];
            acc = wmma_f32_16x16x4(a, b, acc);
        }
        const float bv = bias[nt * 16 + m];
        #pragma unroll
        for (int r = 0; r < 8; ++r) {
            const int row = row0 + mt * 16 + rb + r;
            outp[(size_t)row * HF_ + nt * 16 + m] = acc[r] + bv;
        }
    }
}

// ---------------------------------------------------------------------------
// Kernel 2: scores + edge-softmax + WMMA aggregation + head-mean.
// Grid: B*4 = 256 blocks x 256 threads. Block = (batch b, chunk of 32 dest i).
// Score phase: f-major, 4 destination nodes per wave processed together,
// b128 LDS reads; leaky-relu folded as e += 0.6a*s + 0.4a*|s| (abs = free
// source modifier, pure FMA pipe, no compare/select).
// ---------------------------------------------------------------------------
__global__ __launch_bounds__(256) void gat_attn_kernel(
    const float* __restrict__ fsrc, const float* __restrict__ fdst,
    const float* __restrict__ attn_w, float* __restrict__ out)
{
    __shared__ float fs[128][68];   // fsrc[b, j, h, :]  (all sources)
    __shared__ float fd[32][68];    // fdst rows for this chunk's i's
    __shared__ float aw[64];        // attn_w[h, :]
    __shared__ float al[32][132];   // alpha[i_local][j], padded for A-reads

    const int tid  = threadIdx.x;
    const int wave = tid >> 5;
    const int lane = tid & 31;
    const int b    = blockIdx.x >> 2;
    const int i0   = (blockIdx.x & 3) * 32;

    const int m    = lane & 15;
    const int koff = (lane >> 4) << 1;
    const int rb   = (lane >> 4) << 3;
    const int it   = wave >> 2;          // i-tile within chunk: 0..1
    const int nt   = wave & 3;           // f-tile: 0..3
    const int il0  = wave * 4;           // first of this wave's 4 dest nodes

    v8f acc = {};   // aggregated over heads

    for (int h = 0; h < 2; ++h) {
        // ---- Phase A: stage LDS (+ prefetch head 1 while loading head 0) ----
        for (int v = tid; v < 128 * 16; v += 256) {
            const int j = v >> 4, c4 = (v & 15) << 2;
            const float* src = fsrc + (size_t)(b * W_ + j) * HF_ + h * OUTF_ + c4;
            *(float4*)&fs[j][c4] = *(const float4*)src;
            if (h == 0) __builtin_prefetch(src + OUTF_, 0, 1);
        }
        for (int v = tid; v < 32 * 16; v += 256) {
            const int ii = v >> 4, c4 = (v & 15) << 2;
            const float* src = fdst + (size_t)(b * W_ + i0 + ii) * HF_ + h * OUTF_ + c4;
            *(float4*)&fd[ii][c4] = *(const float4*)src;
            if (h == 0) __builtin_prefetch(src + OUTF_, 0, 1);
        }
        if (tid < 64) aw[tid] = attn_w[h * OUTF_ + tid];
        __syncthreads();

        // ---- Phase B: scores.  lane owns j = lane + 32c, wave owns i = il0..il0+3
        float e[4][4];                        // [dest ii][j group c]
        #pragma unroll
        for (int ii = 0; ii < 4; ++ii)
            #pragma unroll
            for (int c = 0; c < 4; ++c) e[ii][c] = 0.f;

        for (int f4 = 0; f4 < 64; f4 += 4) {
            const float4 a  = *(const float4*)&aw[f4];
            float4 a6, a4v;
            a6.x  = 0.6f * a.x; a6.y  = 0.6f * a.y; a6.z  = 0.6f * a.z; a6.w  = 0.6f * a.w;
            a4v.x = 0.4f * a.x; a4v.y = 0.4f * a.y; a4v.z = 0.4f * a.z; a4v.w = 0.4f * a.w;
            float4 d[4];
            #pragma unroll
            for (int ii = 0; ii < 4; ++ii)
                d[ii] = *(const float4*)&fd[il0 + ii][f4];   // broadcast b128

            #pragma unroll
            for (int c = 0; c < 4; ++c) {
                const float4 s4 = *(const float4*)&fs[lane + 32 * c][f4];
#define LRELU_ACC(comp)                                                        \
                {                                                              \
                    const float a6c = a6.comp, a4c = a4v.comp;                 \
                    _Pragma("unroll")                                          \
                    for (int ii = 0; ii < 4; ++ii) {                           \
                        const float s = s4.comp + d[ii].comp;                  \
                        e[ii][c] = fmaf(a6c, s, e[ii][c]);                     \
                        e[ii][c] = fmaf(a4c, fabsf(s), e[ii][c]);              \
                    }                                                          \
                }
                LRELU_ACC(x) LRELU_ACC(y) LRELU_ACC(z) LRELU_ACC(w)
#undef LRELU_ACC
            }
        }

        // ---- softmax over j per destination node ----
        #pragma unroll
        for (int ii = 0; ii < 4; ++ii) {
            float mx = fmaxf(fmaxf(e[ii][0], e[ii][1]), fmaxf(e[ii][2], e[ii][3]));
            #pragma unroll
            for (int off = 16; off; off >>= 1)
                mx = fmaxf(mx, __shfl_xor(mx, off, 32));
            const float x0 = __expf(e[ii][0] - mx), x1 = __expf(e[ii][1] - mx);
            const float x2 = __expf(e[ii][2] - mx), x3 = __expf(e[ii][3] - mx);
            float sum = (x0 + x1) + (x2 + x3);
            #pragma unroll
            for (int off = 16; off; off >>= 1)
                sum += __shfl_xor(sum, off, 32);
            const float inv = 1.0f / sum;
            al[il0 + ii][lane +  0] = x0 * inv;
            al[il0 + ii][lane + 32] = x1 * inv;
            al[il0 + ii][lane + 64] = x2 * inv;
            al[il0 + ii][lane + 96] = x3 * inv;
        }
        __syncthreads();

        // ---- Phase C: out_tile[16i x 16f] += alpha[16 x 128] @ fs[128 x 16] ----
        #pragma unroll
        for (int k0 = 0; k0 < 128; k0 += 4) {
            v2f a, bb;
            a.x  = al[it * 16 + m][k0 + koff];
            a.y  = al[it * 16 + m][k0 + koff + 1];
            bb.x = fs[k0 + koff][nt * 16 + m];
            bb.y = fs[k0 + koff + 1][nt * 16 + m];
            acc = wmma_f32_16x16x4(a, bb, acc);
        }
        __syncthreads();   // protect fs/fd/al before next head reload
    }

    // ---- head mean (H=2) + store: out[b, i, f] ----
    #pragma unroll
    for (int r = 0; r < 8; ++r) {
        const int i = i0 + it * 16 + rb + r;
        out[((size_t)(b * W_ + i)) * OUTF_ + nt * 16 + m] = 0.5f * acc[r];
    }
}

// ---------------------------------------------------------------------------
extern "C" void kernel_launch(void* const* d_in, const int* in_sizes, int n_in,
                              void* d_out, int out_size, void* d_ws, size_t ws_size,
                              hipStream_t stream) {
    const float* x      = (const float*)d_in[0];
    const float* w_src  = (const float*)d_in[1];
    const float* b_src  = (const float*)d_in[2];
    const float* w_dst  = (const float*)d_in[3];
    const float* b_dst  = (const float*)d_in[4];
    const float* attn_w = (const float*)d_in[5];

    float* fsrc = (float*)d_ws;                      // [8192,128]
    float* fdst = fsrc + (size_t)ROWS_ * HF_;        // [8192,128]

    gat_proj_kernel<<<128, 256, 0, stream>>>(x, w_src, b_src, w_dst, b_dst,
                                             fsrc, fdst);
    gat_attn_kernel<<<B_ * 4, 256, 0, stream>>>(fsrc, fdst, attn_w,
                                                (float*)d_out);
}